// InfoNCELoss_41678362640598
// MI455X (gfx1250) — compile-verified
//
#include <hip/hip_runtime.h>
#include <math.h>

// InfoNCE loss: B=8192, D=128, K=256, T=0.07
#define TEMP_INV (1.0f / 0.07f)
#define DDIM 128
#define KNEG 256

typedef __attribute__((ext_vector_type(2))) float v2f;
typedef __attribute__((ext_vector_type(8))) float v8f;

// ---------------------------------------------------------------------------
// Kernel 1: per-row inverse L2 norms of z1/z2 + positive similarity.
// One 128-thread block per row (4 waves). Stores 1/max(||z||,eps) so the
// WMMA kernel can normalize on the fly (avoids materializing z1n/z2n).
// ---------------------------------------------------------------------------
__global__ __launch_bounds__(128) void rownorm_kernel(
    const float* __restrict__ z1, const float* __restrict__ z2,
    float* __restrict__ invn1, float* __restrict__ invn2,
    float* __restrict__ pos) {
  __shared__ float sa[DDIM], sb[DDIM], sc[DDIM];
  const int b = blockIdx.x;
  const int d = threadIdx.x;
  const float x = z1[b * DDIM + d];
  const float y = z2[b * DDIM + d];
  sa[d] = x * x;
  sb[d] = y * y;
  sc[d] = x * y;
  __syncthreads();
  for (int off = DDIM / 2; off > 0; off >>= 1) {
    if (d < off) {
      sa[d] += sa[d + off];
      sb[d] += sb[d + off];
      sc[d] += sc[d + off];
    }
    __syncthreads();
  }
  if (d == 0) {
    const float i1 = 1.0f / fmaxf(sqrtf(sa[0]), 1e-12f);
    const float i2 = 1.0f / fmaxf(sqrtf(sb[0]), 1e-12f);
    invn1[b] = i1;
    invn2[b] = i2;
    pos[b] = sc[0] * i1 * i2 * TEMP_INV;  // dot(z1n,z2n)/T
  }
}

// ---------------------------------------------------------------------------
// Kernel 2: gathered negative similarities via V_WMMA_F32_16X16X4_F32.
// One wave per (row b, 16-negative tile):
//   A (16x4 slice): 16 gathered z2 rows, scaled by invn2[j]  (per-lane scalar)
//   B (4x16 slice): z1[b] broadcast to all 16 columns, scaled by invn1[b]
//   C accumulates over D in steps of 4 -> 32 WMMAs per wave.
// f32 A 16x4 layout: lane L holds A[M=L%16][K = 2*(L/16) + v], v in {0,1}
// f32 B 4x16 layout: VGPR v, lanes 0-15 -> K=2*(L/16)+v (value indep. of N)
// C/D 16x16: VGPR r, lanes 0-15 -> (M=r, N=lane); lanes 16-31 -> (M=r+8).
// Column N=0 (lanes 0 and 16) carries the 16 negative dots.
// ---------------------------------------------------------------------------
__global__ __launch_bounds__(256) void negdot_wmma_kernel(
    const float* __restrict__ z1, const float* __restrict__ z2,
    const int* __restrict__ idx, const float* __restrict__ invn1,
    const float* __restrict__ invn2, float* __restrict__ negsim) {
  const int wave = blockIdx.x * (256 / 32) + (threadIdx.x >> 5);
  const int lane = threadIdx.x & 31;
  const int b = wave >> 4;             // 16 k-tiles per row
  const int k0 = (wave & 15) * 16;

  const int j = idx[b * KNEG + k0 + (lane & 15)];  // this lane's A-row source
  const float sA = invn2[j];
  const float sB = invn1[b];
  const float* __restrict__ pa = z2 + (long)j * DDIM;
  const float* __restrict__ pb = z1 + (long)b * DDIM;
  const int half = (lane >> 4) * 2;    // K offset 0 (lanes 0-15) or 2 (16-31)

  v8f c = {};
#pragma unroll
  for (int d0 = 0; d0 < DDIM; d0 += 4) {
    v2f a = *(const v2f*)(pa + d0 + half);   // global_load_b64 (L2-resident)
    v2f bb = *(const v2f*)(pb + d0 + half);
    a *= sA;   // fold L2-normalization into operands
    bb *= sB;
    c = __builtin_amdgcn_wmma_f32_16x16x4_f32(
        /*neg_a=*/false, a, /*neg_b=*/false, bb,
        /*c_mod=*/(short)0, c, /*reuse_a=*/false, /*reuse_b=*/false);
  }

  // Lanes 0 and 16 hold output column N=0: M = r (+8 for upper half).
  if ((lane & 15) == 0) {
    float* __restrict__ o = negsim + (long)b * KNEG + k0 + (lane >> 4) * 8;
#pragma unroll
    for (int r = 0; r < 8; ++r) o[r] = c[r] * TEMP_INV;
  }
}

// ---------------------------------------------------------------------------
// Kernel 3: per-row loss = logsumexp(logits) - pos  (label 0 cross-entropy).
// One 256-thread block per row.
// ---------------------------------------------------------------------------
__global__ __launch_bounds__(256) void rowloss_kernel(
    const float* __restrict__ negsim, const float* __restrict__ pos,
    float* __restrict__ rowloss) {
  __shared__ float sm[KNEG];
  const int b = blockIdx.x;
  const int t = threadIdx.x;
  const float v = negsim[(long)b * KNEG + t];
  const float p = pos[b];
  sm[t] = fmaxf(v, p);
  __syncthreads();
  for (int off = KNEG / 2; off > 0; off >>= 1) {
    if (t < off) sm[t] = fmaxf(sm[t], sm[t + off]);
    __syncthreads();
  }
  const float mx = sm[0];
  __syncthreads();
  sm[t] = expf(v - mx) + (t == 0 ? expf(p - mx) : 0.0f);
  __syncthreads();
  for (int off = KNEG / 2; off > 0; off >>= 1) {
    if (t < off) sm[t] += sm[t + off];
    __syncthreads();
  }
  if (t == 0) rowloss[b] = logf(sm[0]) + mx - p;
}

// ---------------------------------------------------------------------------
// Kernel 4: deterministic fixed-order mean over rows (no float atomics).
// ---------------------------------------------------------------------------
__global__ __launch_bounds__(256) void final_kernel(
    const float* __restrict__ rowloss, float* __restrict__ out, int nrows) {
  __shared__ float sm[256];
  const int t = threadIdx.x;
  float s = 0.0f;
  for (int i = t; i < nrows; i += 256) s += rowloss[i];
  sm[t] = s;
  __syncthreads();
  for (int off = 128; off > 0; off >>= 1) {
    if (t < off) sm[t] += sm[t + off];
    __syncthreads();
  }
  if (t == 0) out[0] = sm[0] / (float)nrows;
}

extern "C" void kernel_launch(void* const* d_in, const int* in_sizes, int n_in,
                              void* d_out, int out_size, void* d_ws,
                              size_t ws_size, hipStream_t stream) {
  const float* z1 = (const float*)d_in[0];
  const float* z2 = (const float*)d_in[1];
  const int* idx = (const int*)d_in[2];

  const int B = in_sizes[0] / DDIM;  // 8192

  // Workspace layout (floats): invn1[B] | invn2[B] | pos[B] | rowloss[B] | negsim[B*K]
  float* invn1 = (float*)d_ws;
  float* invn2 = invn1 + B;
  float* pos = invn2 + B;
  float* rowloss = pos + B;
  float* negsim = rowloss + B;

  rownorm_kernel<<<B, DDIM, 0, stream>>>(z1, z2, invn1, invn2, pos);

  const int waves = B * (KNEG / 16);          // one wave per 16-neg tile
  negdot_wmma_kernel<<<waves / 8, 256, 0, stream>>>(z1, z2, idx, invn1, invn2,
                                                    negsim);

  rowloss_kernel<<<B, 256, 0, stream>>>(negsim, pos, rowloss);
  final_kernel<<<1, 256, 0, stream>>>(rowloss, (float*)d_out, B);
}